// DiffusionConvolution_61272003445087
// MI455X (gfx1250) — compile-verified
//
#include <hip/hip_runtime.h>

// Problem constants (match reference setup_inputs()).
#define NB   32      // batch
#define NN   20000   // nodes
#define ND   64      // input features
#define NF   2048    // ND * NB  (row width of x0 / y buffers)
#define NOUT 64      // output features
#define NK   384     // ND * (K+1) * 2  (GEMM reduction dim)

typedef float v2f __attribute__((ext_vector_type(2)));
typedef float v8f __attribute__((ext_vector_type(8)));

// ---------------------------------------------------------------------------
// Kernel 1: x0[n, d*32 + b] = inputs[b, n, d]   (LDS tile transpose per node)
// ---------------------------------------------------------------------------
__global__ void dcrnn_transpose(const float* __restrict__ in, float* __restrict__ x0) {
    __shared__ float tile[NB][ND + 1];   // +1 pad: kill LDS bank conflicts
    const int n = blockIdx.x;
    const int t = threadIdx.x;           // 256 threads
#pragma unroll
    for (int i = 0; i < 8; ++i) {
        int idx = i * 256 + t;           // idx = b*64 + d  -> coalesced 256B runs
        int b = idx >> 6, d = idx & 63;
        tile[b][d] = in[(size_t)b * NN * ND + (size_t)n * ND + d];
    }
    __syncthreads();
#pragma unroll
    for (int i = 0; i < 8; ++i) {
        int idx = i * 256 + t;           // idx = d*32 + b  -> fully coalesced store
        int d = idx >> 5, b = idx & 31;
        x0[(size_t)n * NF + idx] = tile[b][d];
    }
}

// ---------------------------------------------------------------------------
// Kernel 2: zero-fill scratch y buffers (float4 stores)
// ---------------------------------------------------------------------------
__global__ void dcrnn_zero(float4* __restrict__ p, long n4) {
    long i = (long)blockIdx.x * blockDim.x + threadIdx.x;
    if (i < n4) p[i] = make_float4(0.f, 0.f, 0.f, 0.f);
}

// ---------------------------------------------------------------------------
// Kernel 3: SpMM scatter:  y[rows[e], :] += vals[e] * x[cols[e], :]
// One workgroup per edge; 256 lanes x 2 float4 = 2048 floats.
// global_atomic_add_f32 (relaxed, agent scope, no return) -> L2 atomics.
// ---------------------------------------------------------------------------
__global__ void dcrnn_spmm(const int* __restrict__ rows, const int* __restrict__ cols,
                           const float* __restrict__ vals,
                           const float* __restrict__ x, float* __restrict__ y) {
    const int e = blockIdx.x;                  // uniform -> scalar loads
    const int r = rows[e];
    const int c = cols[e];
    const float v = vals[e];
    const float4* xs = (const float4*)(x + (size_t)c * NF);
    float* yd = y + (size_t)r * NF;
    const int t = threadIdx.x;

    float4 a = xs[t];
    float4 b = xs[t + 256];
    int o = 4 * t;
    __hip_atomic_fetch_add(&yd[o + 0], v * a.x, __ATOMIC_RELAXED, __HIP_MEMORY_SCOPE_AGENT);
    __hip_atomic_fetch_add(&yd[o + 1], v * a.y, __ATOMIC_RELAXED, __HIP_MEMORY_SCOPE_AGENT);
    __hip_atomic_fetch_add(&yd[o + 2], v * a.z, __ATOMIC_RELAXED, __HIP_MEMORY_SCOPE_AGENT);
    __hip_atomic_fetch_add(&yd[o + 3], v * a.w, __ATOMIC_RELAXED, __HIP_MEMORY_SCOPE_AGENT);
    o += 1024;
    __hip_atomic_fetch_add(&yd[o + 0], v * b.x, __ATOMIC_RELAXED, __HIP_MEMORY_SCOPE_AGENT);
    __hip_atomic_fetch_add(&yd[o + 1], v * b.y, __ATOMIC_RELAXED, __HIP_MEMORY_SCOPE_AGENT);
    __hip_atomic_fetch_add(&yd[o + 2], v * b.z, __ATOMIC_RELAXED, __HIP_MEMORY_SCOPE_AGENT);
    __hip_atomic_fetch_add(&yd[o + 3], v * b.w, __ATOMIC_RELAXED, __HIP_MEMORY_SCOPE_AGENT);
}

// ---------------------------------------------------------------------------
// Kernel 4: per-node WMMA GEMM.
//   Out_n^T[o=64, b=32] = W^T[64, 384] @ Xtilde_n[384, 32]     (f32 WMMA 16x16x4)
// Block = 256 threads = 8 waves; wave -> (o-tile in 0..3, b-tile in 0..1).
// Fragment layouts per ISA §7.12.2:
//   A 16x4 f32 : lanes 0-15 {v0=K0, v1=K1}, lanes 16-31 {v0=K2, v1=K3}, M=lane%16
//   B  4x16 f32: v0 = {row K0 | row K2}, v1 = {row K1 | row K3}, N=lane%16
//   D 16x16 f32: vgpr i -> M = i (lanes<16) / i+8 (lanes>=16), N = lane%16
// ---------------------------------------------------------------------------
__global__ void dcrnn_gemm(const float* __restrict__ x0,
                           const float* __restrict__ y10, const float* __restrict__ y20,
                           const float* __restrict__ y11, const float* __restrict__ y21,
                           const float* __restrict__ W, const float* __restrict__ bias,
                           float* __restrict__ out) {
    const int n    = blockIdx.x;
    const int wv   = threadIdx.x >> 5;
    const int lane = threadIdx.x & 31;
    const int o0   = (wv & 3) << 4;      // 0,16,32,48
    const int b0   = (wv >> 2) << 4;     // 0,16
    const int l16  = lane & 15;
    const int khi  = (lane >= 16) ? 2 : 0;

    const float* xs[6] = { x0, y10, y20, x0, y11, y21 };

    v8f acc = {0.f, 0.f, 0.f, 0.f, 0.f, 0.f, 0.f, 0.f};

#pragma unroll
    for (int s = 0; s < 6; ++s) {
        const float* Xn = xs[s] + (size_t)n * NF;        // [64 rows d][32 cols b]
        const float* Wf = W + (size_t)(s * ND) * NOUT;   // rows f = s*64 + d
#pragma unroll
        for (int kd = 0; kd < ND; kd += 4) {
            const int ka = kd + khi;     // per-lane K pair base
            v2f a, b;
            // A = W^T : A[M=o, K=f] = W[f, o]; lanes read 64B-contiguous W rows
            a.x = Wf[(ka + 0) * NOUT + o0 + l16];
            a.y = Wf[(ka + 1) * NOUT + o0 + l16];
            // B = Xtilde : B[K=f, N=b] = Xn[d*32 + b]; contiguous 64B half-rows
            b.x = Xn[(ka + 0) * NB + b0 + l16];
            b.y = Xn[(ka + 1) * NB + b0 + l16];
            acc = __builtin_amdgcn_wmma_f32_16x16x4_f32(
                false, a, false, b, (short)0, acc, false, false);
        }
    }

    // D tile: lane half selects o-offset 0/8; 8 consecutive o values per lane.
    const int ob = o0 + ((lane >= 16) ? 8 : 0);
    float4 bv0 = *(const float4*)(bias + ob);
    float4 bv1 = *(const float4*)(bias + ob + 4);
    float* op = out + (size_t)(b0 + l16) * NN * NOUT + (size_t)n * NOUT + ob;
    *(float4*)(op)     = make_float4(acc[0] + bv0.x, acc[1] + bv0.y,
                                     acc[2] + bv0.z, acc[3] + bv0.w);
    *(float4*)(op + 4) = make_float4(acc[4] + bv1.x, acc[5] + bv1.y,
                                     acc[6] + bv1.z, acc[7] + bv1.w);
}

// ---------------------------------------------------------------------------
extern "C" void kernel_launch(void* const* d_in, const int* in_sizes, int n_in,
                              void* d_out, int out_size, void* d_ws, size_t ws_size,
                              hipStream_t stream) {
    const float* inputs = (const float*)d_in[0];
    const int*   r0 = (const int*)d_in[1];
    const int*   c0 = (const int*)d_in[2];
    const float* v0 = (const float*)d_in[3];
    const int*   r1 = (const int*)d_in[4];
    const int*   c1 = (const int*)d_in[5];
    const float* v1 = (const float*)d_in[6];
    const float* W  = (const float*)d_in[7];
    const float* bias = (const float*)d_in[8];
    float* out = (float*)d_out;

    const int E = in_sizes[1];           // 640000 edges per support
    const size_t NNF = (size_t)NN * NF;  // 40.96M floats per buffer

    float* ws  = (float*)d_ws;           // needs 5 * NNF * 4B = 819.2 MB
    float* x0  = ws;
    float* y10 = ws + 1 * NNF;
    float* y20 = ws + 2 * NNF;
    float* y11 = ws + 3 * NNF;
    float* y21 = ws + 4 * NNF;

    // 1) transpose inputs -> x0 [N, D*B]
    dcrnn_transpose<<<NN, 256, 0, stream>>>(inputs, x0);

    // 2) zero the four SpMM destination buffers
    long n4 = (long)(4 * NNF) / 4;       // float4 count
    dcrnn_zero<<<(int)((n4 + 255) / 256), 256, 0, stream>>>((float4*)y10, n4);

    // 3) diffusion steps (stream-ordered: each SpMM sees completed predecessor)
    dcrnn_spmm<<<E, 256, 0, stream>>>(r0, c0, v0, x0,  y10);
    dcrnn_spmm<<<E, 256, 0, stream>>>(r0, c0, v0, y10, y20);
    dcrnn_spmm<<<E, 256, 0, stream>>>(r1, c1, v1, x0,  y11);
    dcrnn_spmm<<<E, 256, 0, stream>>>(r1, c1, v1, y11, y21);

    // 4) fused projection: out[b,n,:] = diff @ W + bias  via f32 WMMA
    dcrnn_gemm<<<NN, 256, 0, stream>>>(x0, y10, y20, y11, y21, W, bias, out);
}